// AssociativeMemoryStep_11373073400439
// MI455X (gfx1250) — compile-verified
//
#include <hip/hip_runtime.h>

// ---------------------------------------------------------------------------
// AssociativeMemoryStep on MI455X (gfx1250): f32 WMMA (16x16x4) pipeline.
// B=4, T=4096, V=1024, C=256, F=2*N_BASIS=256.
// ---------------------------------------------------------------------------

typedef float v2f __attribute__((ext_vector_type(2)));
typedef float v8f __attribute__((ext_vector_type(8)));

#define C_DIM 256
#define V_DIM 1024
#define F_DIM 256
#define T_DIM 4096
#define B_DIM 4
#define M_DIM (B_DIM * T_DIM)   // 16384

static __device__ __forceinline__ v8f wmma4(v2f a, v2f b, v8f c) {
  // D = A(16x4 f32) * B(4x16 f32) + C(16x16 f32)
  return __builtin_amdgcn_wmma_f32_16x16x4_f32(
      /*neg_a=*/false, a, /*neg_b=*/false, b,
      /*c_mod=*/(short)0, c, /*reuse_a=*/false, /*reuse_b=*/false);
}

static __device__ __forceinline__ v8f vzero8() {
  v8f z;
#pragma unroll
  for (int i = 0; i < 8; ++i) z[i] = 0.0f;
  return z;
}

// ---------------------------------------------------------------------------
// Kernel 1: w = basis[V,F] @ coeffs.T  ->  [V,C].  o-weights stored transposed.
// One wave per 16x16 tile. blockIdx.z selects {q,k,v,o}.  (~0.5 GFLOP, cold.)
// ---------------------------------------------------------------------------
__global__ void proj_weights_kernel(const float* __restrict__ basis,
                                    const float* __restrict__ qc,
                                    const float* __restrict__ kc,
                                    const float* __restrict__ vc,
                                    const float* __restrict__ oc,
                                    float* __restrict__ qw,
                                    float* __restrict__ kw,
                                    float* __restrict__ vw,
                                    float* __restrict__ owT) {
  const int lane = threadIdx.x & 31;
  const int r = lane & 15;
  const int h = lane >> 4;
  const int v0 = blockIdx.x * 16;
  const int c0 = blockIdx.y * 16;
  const int which = blockIdx.z;
  const float* coef = (which == 0) ? qc : (which == 1) ? kc : (which == 2) ? vc : oc;

  v8f acc = vzero8();
  for (int f0 = 0; f0 < F_DIM; f0 += 4) {
    // A: basis rows (M=v), K=f.  lane r holds row v0+r; half selects K pair.
    v2f a = *(const v2f*)&basis[(size_t)(v0 + r) * F_DIM + f0 + 2 * h];
    // B[k][n] = coeffs[c0+n][k]  -> contiguous float2 per lane.
    v2f b = *(const v2f*)&coef[(size_t)(c0 + r) * F_DIM + f0 + 2 * h];
    acc = wmma4(a, b, acc);
  }
  if (which < 3) {
    float* w = (which == 0) ? qw : (which == 1) ? kw : vw;
#pragma unroll
    for (int i = 0; i < 8; ++i)
      w[(size_t)(v0 + i + 8 * h) * C_DIM + c0 + r] = acc[i];
  } else {
#pragma unroll
    for (int i = 0; i < 8; ++i)
      owT[(size_t)(c0 + r) * V_DIM + v0 + i + 8 * h] = acc[i];
  }
}

// ---------------------------------------------------------------------------
// Kernel 2: {Q,K,V}[M,C] = x[M,V] @ w[V,C].
// One wave per 16x64 tile: A-fragment reused across 4 WMMAs (register
// N-blocking) to raise the wmma:vmem issue ratio.
// ---------------------------------------------------------------------------
__global__ void qkv_kernel(const float* __restrict__ x,
                           const float* __restrict__ qw,
                           const float* __restrict__ kw,
                           const float* __restrict__ vw,
                           float* __restrict__ Q,
                           float* __restrict__ K,
                           float* __restrict__ Vv) {
  const int lane = threadIdx.x & 31;
  const int r = lane & 15;
  const int h = lane >> 4;
  const int m0 = blockIdx.x * 16;
  const int c0 = blockIdx.y * 64;
  const int which = blockIdx.z;
  const float* w = (which == 0) ? qw : (which == 1) ? kw : vw;
  float* out = (which == 0) ? Q : (which == 1) ? K : Vv;

  v8f acc[4];
#pragma unroll
  for (int nt = 0; nt < 4; ++nt) acc[nt] = vzero8();

  for (int k0 = 0; k0 < V_DIM; k0 += 4) {
    v2f a = *(const v2f*)&x[(size_t)(m0 + r) * V_DIM + k0 + 2 * h];
    const float* w0 = &w[(size_t)(k0 + 2 * h) * C_DIM + c0 + r];
    const float* w1 = w0 + C_DIM;
#pragma unroll
    for (int nt = 0; nt < 4; ++nt) {
      v2f b;  // B[k][n] = w[k][c0+n] : coalesced across lanes
      b.x = w0[nt * 16];
      b.y = w1[nt * 16];
      acc[nt] = wmma4(a, b, acc[nt]);
    }
  }
#pragma unroll
  for (int nt = 0; nt < 4; ++nt)
#pragma unroll
    for (int i = 0; i < 8; ++i)
      out[(size_t)(m0 + i + 8 * h) * C_DIM + c0 + nt * 16 + r] = acc[nt][i];
}

// ---------------------------------------------------------------------------
// Kernel 3: retrieved = (Q K^T * decay-mask) V, per batch.
// One wave -> 16 rows x 64 channels of retrieved. Q tile staged in LDS
// (padded stride 260 -> conflict-free b64 column reads). Score tile is
// round-tripped through LDS (pad 18) to re-layout C-fragment -> A-fragment.
// Mask: weight(i,j) = decay^max(j-i-1,0) for j>i else 0 (upper-triangular).
// Next j-tile's K/V rows prefetched (global_prefetch_b8) behind the WMMAs.
// ---------------------------------------------------------------------------
__global__ void attn_kernel(const float* __restrict__ Q,
                            const float* __restrict__ K,
                            const float* __restrict__ Vv,
                            const float* __restrict__ decay_logit,
                            float* __restrict__ R) {
  __shared__ float qs[16 * 260];
  __shared__ float ss[16 * 18];

  const int lane = threadIdx.x & 31;
  const int r = lane & 15;
  const int h = lane >> 4;
  const int i0 = blockIdx.x * 16;
  const int c0 = blockIdx.y * 64;
  const int b = blockIdx.z;

  const float* Qb = Q + (size_t)b * T_DIM * C_DIM;
  const float* Kb = K + (size_t)b * T_DIM * C_DIM;
  const float* Vb = Vv + (size_t)b * T_DIM * C_DIM;
  float* Rb = R + (size_t)b * T_DIM * C_DIM;

  // Stage the 16x256 Q tile into LDS (float4 granularity, coalesced).
  for (int idx = lane; idx < 16 * 64; idx += 32) {
    int row = idx >> 6;
    int c4 = (idx & 63) << 2;
    *(float4*)&qs[row * 260 + c4] =
        *(const float4*)&Qb[(size_t)(i0 + row) * C_DIM + c4];
  }
  __syncthreads();

  const float dl = *decay_logit;
  const float dec = 1.0f / (1.0f + __expf(-dl));
  const float ln_dec = __logf(dec);

  v8f acc0 = vzero8(), acc1 = vzero8(), acc2 = vzero8(), acc3 = vzero8();

  for (int j0 = i0; j0 < T_DIM; j0 += 16) {
    // Prefetch next tile's K/V rows into cache while this tile computes.
    {
      int jn = (j0 + 16 < T_DIM) ? (j0 + 16) : j0;
      __builtin_prefetch(&Kb[(size_t)(jn + r + 8 * h) * C_DIM], 0, 1);
      __builtin_prefetch(&Vb[(size_t)(jn + r + 8 * h) * C_DIM + c0], 0, 1);
    }

    // scores tile s[16x16] = Q_i (16xC) * K_j^T (Cx16)
    v8f s = vzero8();
    const float* Kj = Kb + (size_t)j0 * C_DIM;
    for (int k0 = 0; k0 < C_DIM; k0 += 4) {
      v2f a = *(const v2f*)&qs[r * 260 + k0 + 2 * h];
      // B[k][n] = K[j0+n][k] : contiguous float2 per lane
      v2f bb = *(const v2f*)&Kj[(size_t)r * C_DIM + k0 + 2 * h];
      s = wmma4(a, bb, s);
    }
    // Apply decay mask, spill to LDS in row-major for A-fragment reload.
#pragma unroll
    for (int i = 0; i < 8; ++i) {
      int gi = i0 + i + 8 * h;   // global row (query index)
      int gj = j0 + r;           // global col (key index)
      int d = gj - gi;
      float wgt = (d > 0) ? __expf(ln_dec * (float)(d - 1)) : 0.0f;
      ss[(i + 8 * h) * 18 + r] = s[i] * wgt;
    }
    __syncthreads();
    // retrieved += s (16x16) * V_j (16x64)
#pragma unroll
    for (int kk = 0; kk < 4; ++kk) {
      v2f a = *(const v2f*)&ss[r * 18 + kk * 4 + 2 * h];
      const float* Vr0 = Vb + (size_t)(j0 + kk * 4 + 2 * h) * C_DIM + c0;
      const float* Vr1 = Vr0 + C_DIM;
      v2f bb;
      bb.x = Vr0[r];      bb.y = Vr1[r];      acc0 = wmma4(a, bb, acc0);
      bb.x = Vr0[16 + r]; bb.y = Vr1[16 + r]; acc1 = wmma4(a, bb, acc1);
      bb.x = Vr0[32 + r]; bb.y = Vr1[32 + r]; acc2 = wmma4(a, bb, acc2);
      bb.x = Vr0[48 + r]; bb.y = Vr1[48 + r]; acc3 = wmma4(a, bb, acc3);
    }
    __syncthreads();
  }

#pragma unroll
  for (int i = 0; i < 8; ++i) {
    size_t row = (size_t)(i0 + i + 8 * h) * C_DIM + c0 + r;
    Rb[row] = acc0[i];
    Rb[row + 16] = acc1[i];
    Rb[row + 32] = acc2[i];
    Rb[row + 48] = acc3[i];
  }
}

// ---------------------------------------------------------------------------
// Kernel 4: out[M,V] = retrieved[M,C] @ owT[C,V] * out_scale.
// One wave per 16x64 tile (register N-blocking, A-fragment reused 4x).
// ---------------------------------------------------------------------------
__global__ void out_proj_kernel(const float* __restrict__ R,
                                const float* __restrict__ owT,
                                const float* __restrict__ out_scale,
                                float* __restrict__ out) {
  const int lane = threadIdx.x & 31;
  const int r = lane & 15;
  const int h = lane >> 4;
  const int m0 = blockIdx.x * 16;
  const int v0 = blockIdx.y * 64;

  v8f acc[4];
#pragma unroll
  for (int nt = 0; nt < 4; ++nt) acc[nt] = vzero8();

  for (int k0 = 0; k0 < C_DIM; k0 += 4) {
    v2f a = *(const v2f*)&R[(size_t)(m0 + r) * C_DIM + k0 + 2 * h];
    const float* w0 = &owT[(size_t)(k0 + 2 * h) * V_DIM + v0 + r];
    const float* w1 = w0 + V_DIM;
#pragma unroll
    for (int nt = 0; nt < 4; ++nt) {
      v2f b;  // B[k][n] = owT[k][v0+n] : coalesced across lanes
      b.x = w0[nt * 16];
      b.y = w1[nt * 16];
      acc[nt] = wmma4(a, b, acc[nt]);
    }
  }
  const float sc = *out_scale;
#pragma unroll
  for (int nt = 0; nt < 4; ++nt)
#pragma unroll
    for (int i = 0; i < 8; ++i)
      out[(size_t)(m0 + i + 8 * h) * V_DIM + v0 + nt * 16 + r] = acc[nt][i] * sc;
}

// ---------------------------------------------------------------------------
extern "C" void kernel_launch(void* const* d_in, const int* in_sizes, int n_in,
                              void* d_out, int out_size, void* d_ws, size_t ws_size,
                              hipStream_t stream) {
  (void)in_sizes; (void)n_in; (void)out_size; (void)ws_size;

  const float* x = (const float*)d_in[0];            // [B,T,V]
  const float* basis = (const float*)d_in[1];        // [V, 2*NB]
  const float* qc = (const float*)d_in[2];           // [C, 2*NB]
  const float* kc = (const float*)d_in[3];
  const float* vc = (const float*)d_in[4];
  const float* oc = (const float*)d_in[5];
  const float* decay_logit = (const float*)d_in[6];  // scalar
  const float* out_scale = (const float*)d_in[7];    // scalar
  float* out = (float*)d_out;                        // [B,T,V]

  // Workspace layout (floats): ~71.3 MB total.
  float* ws = (float*)d_ws;
  float* qw = ws;                                    // [V,C]
  float* kw = qw + (size_t)V_DIM * C_DIM;            // [V,C]
  float* vw = kw + (size_t)V_DIM * C_DIM;            // [V,C]
  float* owT = vw + (size_t)V_DIM * C_DIM;           // [C,V] (transposed)
  float* Q = owT + (size_t)V_DIM * C_DIM;            // [M,C]
  float* K = Q + (size_t)M_DIM * C_DIM;              // [M,C]
  float* Vv = K + (size_t)M_DIM * C_DIM;             // [M,C]
  float* R = Vv + (size_t)M_DIM * C_DIM;             // [M,C]

  dim3 blk(32);

  proj_weights_kernel<<<dim3(V_DIM / 16, C_DIM / 16, 4), blk, 0, stream>>>(
      basis, qc, kc, vc, oc, qw, kw, vw, owT);

  qkv_kernel<<<dim3(M_DIM / 16, C_DIM / 64, 3), blk, 0, stream>>>(
      x, qw, kw, vw, Q, K, Vv);

  attn_kernel<<<dim3(T_DIM / 16, C_DIM / 64, B_DIM), blk, 0, stream>>>(
      Q, K, Vv, decay_logit, R);

  out_proj_kernel<<<dim3(M_DIM / 16, V_DIM / 64), blk, 0, stream>>>(
      R, owT, out_scale, out);
}